// ScaledDotProductAttention_15968688407214
// MI455X (gfx1250) — compile-verified
//
#include <hip/hip_runtime.h>
#include <hip/hip_bf16.h>

typedef float v2f __attribute__((ext_vector_type(2)));
typedef float v4f __attribute__((ext_vector_type(4)));
typedef float v8f __attribute__((ext_vector_type(8)));

// Problem constants (B=4, H=16, S=2048, D=64)
#define ATT_S 2048
#define ATT_D 64
#define ATT_BH 64            // B*H
#define QS_STRIDE 68         // 64 + 4 pad (bank-conflict avoidance)
#define SS_STRIDE 2052       // 2048 + 4 pad -> bank stride 4 for column reads

__global__ __launch_bounds__(256, 1)
void sdpa_f32_wmma_kernel(const float* __restrict__ Q,
                          const float* __restrict__ K,
                          const float* __restrict__ V,
                          const unsigned char* __restrict__ mask,
                          float* __restrict__ out,
                          float* __restrict__ wts) {
    extern __shared__ float smem[];
    float* sQ   = smem;                     // [16][QS_STRIDE]      = 1088 floats
    float* sInv = smem;                     // reuses sQ space after phase 1
    float* pC   = smem + 16 * QS_STRIDE;    // [8][16][16] partials = 2048 floats
    float* sS   = pC + 8 * 256;             // [16][SS_STRIDE]      = 32832 floats

    const int tid = threadIdx.x;
    const int w   = tid >> 5;               // wave id 0..7 (wave32)
    const int l   = tid & 31;               // lane id
    const int lo  = l & 15;                 // lane % 16
    const int hi  = l >> 4;                 // lane / 16

    const int bh = blockIdx.x >> 7;         // 128 query-blocks per (b,h)
    const int q0 = (blockIdx.x & 127) << 4; // first query row of this tile

    const size_t qkvBase = (size_t)bh * ATT_S * ATT_D;
    const size_t mBase   = (size_t)bh * ATT_S * ATT_S + (size_t)q0 * ATT_S;

    // ---- Stage Q tile [16][64] into LDS (coalesced) ----
    for (int i = tid; i < 16 * ATT_D; i += 256) {
        sQ[(i >> 6) * QS_STRIDE + (i & 63)] = Q[qkvBase + (size_t)q0 * ATT_D + i];
    }
    __syncthreads();

    // ---- A-fragments of Q tile (16 K-steps of 4, fp32 WMMA layout) ----
    // A[lane%16][(lane/16)*2 + v]  -> 2 consecutive floats per fragment
    v2f aQ[16];
#pragma unroll
    for (int kk = 0; kk < 16; ++kk) {
        aQ[kk] = *(const v2f*)&sQ[lo * QS_STRIDE + kk * 4 + hi * 2];
    }
    __syncthreads();   // sQ space is reused as sInv later

    // ---- Phase 1: scores = mask ? -1e9 : (Q K^T)/8, kept in LDS ----
    // Each wave owns a CONTIGUOUS 256-key span so mask cachelines are
    // consumed by a single wave across consecutive iterations.
    const float scale = 0.125f; // 1/sqrt(64)
    for (int kt = (w << 4); kt < (w << 4) + 16; ++kt) {
        const int kb = kt << 4;
        v8f c = {};
        // B-fragment: B[(lane/16)*2 + v][lane%16] = K[kb + lane%16][kk*4 + (lane/16)*2 + v]
        const float* kp = K + qkvBase + (size_t)(kb + lo) * ATT_D + hi * 2;
#pragma unroll
        for (int kk = 0; kk < 16; ++kk) {
            v2f b = *(const v2f*)(kp + kk * 4);
            c = __builtin_amdgcn_wmma_f32_16x16x4_f32(
                    false, aQ[kk], false, b, (short)0, c, false, false);
        }
        // One base pointer; per-row offsets are compile-time immediates (r*2048)
        const unsigned char* mp = mask + mBase + kb + lo + (size_t)hi * 8 * ATT_S;
#pragma unroll
        for (int r = 0; r < 8; ++r) {
            const int M = r + (hi << 3);
            const unsigned char mv = __builtin_nontemporal_load(&mp[r * ATT_S]);
            const float s = mv ? -1.0e9f : c[r] * scale;
            sS[M * SS_STRIDE + kb + lo] = s;
        }
    }
    __syncthreads();

    // ---- Phase 2: row softmax, b128-vectorized. LDS keeps UNNORMALIZED e;
    //      weights stream = e * inv (NT); inv folded into output epilogue. ----
    for (int rr = 0; rr < 2; ++rr) {
        const int row = (w << 1) + rr;
        float* srow = &sS[row * SS_STRIDE];

        float m = -3.4e38f;
        for (int j = (l << 2); j < ATT_S; j += 128) {
            v4f x = *(const v4f*)&srow[j];
            m = fmaxf(m, fmaxf(fmaxf(x.x, x.y), fmaxf(x.z, x.w)));
        }
#pragma unroll
        for (int off = 16; off; off >>= 1) m = fmaxf(m, __shfl_xor(m, off, 32));

        float sum = 0.0f;
        for (int j = (l << 2); j < ATT_S; j += 128) {
            v4f x = *(const v4f*)&srow[j];
            v4f e;
            e.x = __expf(x.x - m); e.y = __expf(x.y - m);
            e.z = __expf(x.z - m); e.w = __expf(x.w - m);
            *(v4f*)&srow[j] = e;
            sum += (e.x + e.y) + (e.z + e.w);
        }
#pragma unroll
        for (int off = 16; off; off >>= 1) sum += __shfl_xor(sum, off, 32);

        const float inv = 1.0f / sum;
        if (l == 0) sInv[row] = inv;

        float* wrow = wts + mBase + (size_t)row * ATT_S;
        for (int j = (l << 2); j < ATT_S; j += 128) {
            v4f e = *(const v4f*)&srow[j];
            v4f p = { e.x * inv, e.y * inv, e.z * inv, e.w * inv };
            __builtin_nontemporal_store(p, (v4f*)&wrow[j]);   // 1 GB stream
        }
    }
    __syncthreads();

    // ---- Phase 3: O = (e V) * inv via fp32 WMMA. 4 N-tiles x 2 K-halves ----
    {
        const int tile  = w & 3;          // which 16-wide slice of D
        const int khalf = w >> 2;         // which half of K=2048
        const int d0    = tile << 4;
        v8f c = {};
        const float* vp   = V + qkvBase + d0 + lo;        // + k*D per step
        const float* prow = &sS[lo * SS_STRIDE + hi * 2]; // A-fragment base
        const int k0b = khalf * 1024;
#pragma unroll 4
        for (int k0 = k0b; k0 < k0b + 1024; k0 += 4) {
            v2f a = *(const v2f*)(prow + k0);
            v2f b;
            b.x = vp[(size_t)(k0 + hi * 2) * ATT_D];
            b.y = vp[(size_t)(k0 + hi * 2 + 1) * ATT_D];
            c = __builtin_amdgcn_wmma_f32_16x16x4_f32(
                    false, a, false, b, (short)0, c, false, false);
        }
#pragma unroll
        for (int r = 0; r < 8; ++r) {
            pC[w * 256 + (r + (hi << 3)) * 16 + lo] = c[r];
        }
    }
    __syncthreads();

    // ---- Combine K-half partials, apply 1/rowsum, store output tile ----
    for (int i = tid; i < 16 * ATT_D; i += 256) {
        const int row  = i >> 6;
        const int col  = i & 63;
        const int tile = col >> 4;
        const int n    = col & 15;
        const float val = (pC[tile * 256 + row * 16 + n] +
                           pC[(tile + 4) * 256 + row * 16 + n]) * sInv[row];
        out[qkvBase + (size_t)q0 * ATT_D + i] = val;
    }
}

extern "C" void kernel_launch(void* const* d_in, const int* in_sizes, int n_in,
                              void* d_out, int out_size, void* d_ws, size_t ws_size,
                              hipStream_t stream) {
    const float*         q    = (const float*)d_in[0];
    const float*         k    = (const float*)d_in[1];
    const float*         v    = (const float*)d_in[2];
    const unsigned char* mask = (const unsigned char*)d_in[3];

    float* out = (float*)d_out;                                   // [B,H,S,D]
    float* wts = out + (size_t)ATT_BH * ATT_S * ATT_D;            // [B,H,S,S]

    const size_t shmem =
        (size_t)(16 * QS_STRIDE + 8 * 256 + 16 * SS_STRIDE) * sizeof(float); // ~140.5 KB

    (void)hipFuncSetAttribute((const void*)sdpa_f32_wmma_kernel,
                              hipFuncAttributeMaxDynamicSharedMemorySize,
                              (int)shmem);

    const int grid = ATT_BH * (ATT_S / 16); // 8192 workgroups
    sdpa_f32_wmma_kernel<<<grid, 256, shmem, stream>>>(q, k, v, mask, out, wts);
}